// LossNegSampling_70394513981667
// MI455X (gfx1250) — compile-verified
//
#include <hip/hip_runtime.h>
#include <math.h>

#define B_ROWS 16384
#define K_NEG  20
#define D_DIM  128
#define L_COM  64

typedef __attribute__((ext_vector_type(2))) float v2f;
typedef __attribute__((ext_vector_type(8))) float v8f;

__device__ __forceinline__ float log_sigmoid(float x) {
    // numerically stable: min(x,0) - log1p(exp(-|x|))
    return fminf(x, 0.0f) - log1pf(expf(-fabsf(x)));
}

// ---------------- init: zero the two accumulators every call ----------------
__global__ void init_kernel(float* ws) {
    if (threadIdx.x < 2) ws[threadIdx.x] = 0.0f;
}

// ---------------- ||c_l||^2 for the 64 cluster centers ----------------
__global__ void cnorm_kernel(const float* __restrict__ emb_com,
                             float* __restrict__ cnorm) {
    __shared__ float red[128];
    const int n = blockIdx.x;
    const float v = emb_com[n * D_DIM + threadIdx.x];
    red[threadIdx.x] = v * v;
    __syncthreads();
    for (int s = 64; s > 0; s >>= 1) {
        if (threadIdx.x < s) red[threadIdx.x] += red[threadIdx.x + s];
        __syncthreads();
    }
    if (threadIdx.x == 0) cnorm[n] = red[0];
}

// ---------------- negative-sampling loss: one wave32 per row ----------------
// Memory-bound part: 22 gathered 512B rows per b. Each lane loads a float4
// (32 lanes x 16B = 128 floats) -> fully coalesced global_load_b128.
__global__ void score_kernel(const int* __restrict__ u_node,
                             const int* __restrict__ v_node,
                             const int* __restrict__ negs,
                             const float* __restrict__ emb_u,
                             float* __restrict__ loss_acc) {
    __shared__ float wsum[8];
    const int tid  = threadIdx.x;
    const int wave = tid >> 5;
    const int lane = tid & 31;
    const int b    = blockIdx.x * 8 + wave;

    const size_t iu = (size_t)u_node[b];
    const size_t iv = (size_t)v_node[b];
    const float4 u4 = ((const float4*)(emb_u + iu * D_DIM))[lane];
    const float4 v4 = ((const float4*)(emb_u + iv * D_DIM))[lane];

    float pos = u4.x * v4.x + u4.y * v4.y + u4.z * v4.z + u4.w * v4.w;

    float nacc = 0.0f;
    #pragma unroll 4
    for (int k = 0; k < K_NEG; ++k) {
        const size_t in_ = (size_t)negs[b * K_NEG + k];
        const float4 n4  = ((const float4*)(emb_u + in_ * D_DIM))[lane];
        nacc += u4.x * n4.x + u4.y * n4.y + u4.z * n4.z + u4.w * n4.w;
    }

    // wave32 reductions
    for (int o = 16; o > 0; o >>= 1) {
        pos  += __shfl_xor(pos,  o, 32);
        nacc += __shfl_xor(nacc, o, 32);
    }

    if (lane == 0) {
        // neg_score = sum_k (-emb_k) . u = -nacc
        const float s = log_sigmoid(pos) + log_sigmoid(-nacc);
        wsum[wave] = -s;   // loss = -mean(sum_all)
    }
    __syncthreads();
    if (tid == 0) {
        float t = 0.0f;
        #pragma unroll
        for (int w = 0; w < 8; ++w) t += wsum[w];
        atomicAdd(loss_acc, t);
    }
}

// ---------------- clustering: 16 rows per block, WMMA f32 16x16x4 -----------
// S = U(16x128) * C^T(128x64); dist2 = |u|^2 + |c|^2 - 2S.
// 4 waves, wave w owns N-tile w (centers w*16..w*16+15): 32 WMMA K-steps each.
__global__ void cluster_kernel(const int* __restrict__ u_node,
                               const float* __restrict__ emb_u,
                               const float* __restrict__ emb_com,
                               const float* __restrict__ cnorm,
                               float* __restrict__ out_choice,   // d_out + 1
                               float* __restrict__ loss2_acc) {
    __shared__ float lds_u[16 * D_DIM];       // 8 KB
    __shared__ float lds_c[L_COM * D_DIM];    // 32 KB
    __shared__ float lds_un[16];
    __shared__ float lds_cn[L_COM];
    __shared__ int   lds_iu[16];
    __shared__ float wmin[4][16];
    __shared__ int   widx[4][16];

    const int tid  = threadIdx.x;             // 0..127
    const int wave = tid >> 5;
    const int lane = tid & 31;
    const int b0   = blockIdx.x * 16;

    if (tid < 16) { lds_iu[tid] = u_node[b0 + tid]; lds_un[tid] = 0.0f; }
    if (tid < L_COM) lds_cn[tid] = cnorm[tid];
    __syncthreads();

    // stage 16 u-rows (coalesced per row)
    for (int i = tid; i < 16 * D_DIM; i += 128) {
        const int row = i >> 7, col = i & 127;
        lds_u[i] = emb_u[(size_t)lds_iu[row] * D_DIM + col];
    }
    // stage full emb_com (L2-resident, 32 KB)
    for (int i = tid; i < L_COM * D_DIM; i += 128) lds_c[i] = emb_com[i];
    __syncthreads();

    // ||u_m||^2 : thread t does row t/8, 16-column segment t%8
    {
        const int row = tid >> 3, seg = tid & 7;
        float p = 0.0f;
        #pragma unroll
        for (int c = 0; c < 16; ++c) {
            const float x = lds_u[row * D_DIM + seg * 16 + c];
            p += x * x;
        }
        atomicAdd(&lds_un[row], p);
    }
    __syncthreads();

    // WMMA fragments (ISA 7.12.2, 32-bit A 16x4 / B 4x16):
    //   A: lane -> m = lane%16, k = kk + 2*(lane/16) + j   (j = VGPR 0/1)
    //   B: lane -> n = lane%16, k = kk + 2*(lane/16) + j
    // -> both fragments are consecutive float2 LDS loads.
    const int m    = lane & 15;
    const int half = lane >> 4;
    const int nglob = wave * 16 + m;
    const float* arow = &lds_u[m * D_DIM];
    const float* brow = &lds_c[nglob * D_DIM];

    v8f acc = {};
    #pragma unroll 8
    for (int kk = 0; kk < D_DIM; kk += 4) {
        const v2f a  = *(const v2f*)(arow + kk + half * 2);
        const v2f bb = *(const v2f*)(brow + kk + half * 2);
        acc = __builtin_amdgcn_wmma_f32_16x16x4_f32(
                  false, a, false, bb, (short)0, acc, false, false);
    }

    // D layout: VGPR j, lane -> (M = j + 8*half, N = lane%16).
    // min/argmin over N within this 16-wide tile via shfl_xor (stays in half).
    const float cn = lds_cn[nglob];
    float bestv[8]; int besti[8];
    #pragma unroll
    for (int j = 0; j < 8; ++j) {
        const int mrow = j + 8 * half;
        float v = lds_un[mrow] + cn - 2.0f * acc[j];
        int   id = nglob;
        for (int o = 1; o < 16; o <<= 1) {
            const float ov = __shfl_xor(v, o, 32);
            const int   oi = __shfl_xor(id, o, 32);
            if (ov < v || (ov == v && oi < id)) { v = ov; id = oi; }
        }
        bestv[j] = v; besti[j] = id;
    }
    if ((lane & 15) == 0) {
        #pragma unroll
        for (int j = 0; j < 8; ++j) {
            const int mrow = j + 8 * half;
            wmin[wave][mrow] = bestv[j];
            widx[wave][mrow] = besti[j];
        }
    }
    __syncthreads();

    // combine the 4 N-tiles; smaller index wins ties (matches jnp.argmin)
    if (tid < 16) {
        float v = wmin[0][tid]; int id = widx[0][tid];
        #pragma unroll
        for (int w = 1; w < 4; ++w) {
            const float ov = wmin[w][tid]; const int oi = widx[w][tid];
            if (ov < v || (ov == v && oi < id)) { v = ov; id = oi; }
        }
        out_choice[b0 + tid] = (float)id;          // indices 0..63 exact in f32
        atomicAdd(loss2_acc, fmaxf(v, 0.0f));      // min(dist)^2 == min(dist^2)
    }
}

// ---------------- finalize ----------------
__global__ void finalize_kernel(const float* __restrict__ ws,
                                float* __restrict__ out, float gamma) {
    if (threadIdx.x == 0) {
        const float loss  = ws[0] / (float)B_ROWS;
        const float loss2 = ws[1] / (float)B_ROWS;
        out[0] = loss + gamma * loss2;
    }
}

extern "C" void kernel_launch(void* const* d_in, const int* in_sizes, int n_in,
                              void* d_out, int out_size, void* d_ws, size_t ws_size,
                              hipStream_t stream) {
    const int*   u_node  = (const int*)d_in[0];
    const int*   v_node  = (const int*)d_in[1];
    const int*   negs    = (const int*)d_in[2];
    // d_in[3] = nb_labels (scalar 64), known at compile time
    const float* emb_u   = (const float*)d_in[4];
    const float* emb_com = (const float*)d_in[5];

    float* out = (float*)d_out;     // [0]=final_loss, [1..16384]=cluster_choice
    float* ws  = (float*)d_ws;      // [0]=loss_acc, [1]=loss2_acc, [2..65]=cnorm2

    init_kernel<<<1, 32, 0, stream>>>(ws);
    cnorm_kernel<<<L_COM, 128, 0, stream>>>(emb_com, ws + 2);
    score_kernel<<<B_ROWS / 8, 256, 0, stream>>>(u_node, v_node, negs, emb_u, ws);
    cluster_kernel<<<B_ROWS / 16, 128, 0, stream>>>(u_node, emb_u, emb_com,
                                                    ws + 2, out + 1, ws + 1);

    // gamma = 0.01 * 10^(-T_STEP*log10(0.01)/(WALK_L*CTX_W*V*N_SEQ)) = 0.01*10^(4e-9)
    const float gamma = (float)(0.01 * pow(10.0, 2.0 / 5.0e8));
    finalize_kernel<<<1, 32, 0, stream>>>(ws, out, gamma);
}